// TruthGPTMemorySystem_8555574854219
// MI455X (gfx1250) — compile-verified
//
#include <hip/hip_runtime.h>
#include <hip/hip_bf16.h>

#define DIM 1024
#define NKEYS 32768
#define TOPK 10

typedef __attribute__((ext_vector_type(2))) float v2f;
typedef __attribute__((ext_vector_type(8))) float v8f;

// ---------------------------------------------------------------------------
// Kernel 1: q = Wq @ query + bq   (one wave32 per output row)
// ---------------------------------------------------------------------------
__global__ __launch_bounds__(256) void proj_kernel(
    const float* __restrict__ Wq, const float* __restrict__ query,
    const float* __restrict__ bq, float* __restrict__ qout) {
  int lane = threadIdx.x & 31;
  int row  = (blockIdx.x * blockDim.x + threadIdx.x) >> 5;  // 0..1023
  const float* w = Wq + (size_t)row * DIM;
  float s = 0.f;
#pragma unroll 8
  for (int j = lane; j < DIM; j += 32) s = fmaf(w[j], query[j], s);
#pragma unroll
  for (int o = 16; o > 0; o >>= 1) s += __shfl_down(s, o, 32);
  if (lane == 0) qout[row] = s + bq[row];
}

// ---------------------------------------------------------------------------
// Kernel 2: partial[kc][n] = dot(keys[n][k0:k0+klen], q[k0:k0+klen])
// via V_WMMA_F32_16X16X4_F32. A (16x4) = q chunk replicated across M rows;
// B (4x16) = 16 key rows. Lanes 0-15 hold K=0,1; lanes 16-31 hold K=2,3.
// Row M=0 of the accumulator (VGPR0, lanes 0-15) = the 16 partial scores.
// K is split into (1<<lg) chunks across waves for memory-level parallelism:
// up to 8192 waves x 8-load clauses ~= 16 MB in flight to saturate HBM.
// ---------------------------------------------------------------------------
__global__ __launch_bounds__(256) void scores_wmma_kernel(
    const float* __restrict__ keys, const float* __restrict__ qvec,
    float* __restrict__ partial, int lg) {
  __shared__ float qs[DIM];
  for (int i = threadIdx.x; i < DIM; i += blockDim.x) qs[i] = qvec[i];
  __syncthreads();

  int lane  = threadIdx.x & 31;
  int wv    = (blockIdx.x * blockDim.x + threadIdx.x) >> 5;
  int kc    = wv & ((1 << lg) - 1);     // K-chunk id
  int tile  = wv >> lg;                 // 16-row tile id
  int row0  = tile << 4;
  int r     = row0 + (lane & 15);       // this lane's key row (B column)
  int khalf = (lane >> 4) << 1;         // lanes 0-15 -> K 0,1 ; 16-31 -> K 2,3
  int klen  = DIM >> lg;
  int k0    = kc * klen;

  const float* krow = keys + (size_t)r * DIM + k0 + khalf;
  const float* qp   = qs + k0 + khalf;

  v8f acc = {};
#pragma unroll 8
  for (int kb = 0; kb < klen; kb += 4) {
    v2f b = *(const v2f*)(krow + kb);   // B[khalf..khalf+1][lane&15]
    v2f a;                              // A rows identical: q chunk
    a.x = qp[kb];
    a.y = qp[kb + 1];
    acc = __builtin_amdgcn_wmma_f32_16x16x4_f32(
        /*neg_a=*/false, a, /*neg_b=*/false, b,
        /*c_mod=*/(short)0, acc, /*reuse_a=*/false, /*reuse_b=*/false);
  }
  if (lane < 16) partial[(size_t)kc * NKEYS + row0 + lane] = acc[0];  // C[0][lane]
}

// ---------------------------------------------------------------------------
// Kernel 3: single block. Combines K-chunk partials in FIXED order
// (deterministic), 10x exclusion-argmax, softmax denominator, gather.
// ---------------------------------------------------------------------------
__device__ __forceinline__ float score_at(const float* __restrict__ p, int n,
                                          int nchunks) {
  float s = p[n];
  for (int c = 1; c < nchunks; ++c) s += p[(size_t)c * NKEYS + n];
  return s;
}

__global__ __launch_bounds__(1024) void topk_softmax_gather_kernel(
    const float* __restrict__ partial, int nchunks,
    const float* __restrict__ values, float* __restrict__ out) {
  const float inv = 0.03125f;  // 1/sqrt(1024)
  __shared__ float rv[32];
  __shared__ int   ri[32];
  __shared__ float psum[32];
  __shared__ float chosen_s[TOPK];
  __shared__ int   chosen_i[TOPK];
  __shared__ float s_sum;

  int tid  = threadIdx.x;
  int lane = tid & 31;
  int warp = tid >> 5;
  int nw   = blockDim.x >> 5;  // 32 warps

  for (int j = 0; j < TOPK; ++j) {
    float bv = -3.4e38f;
    int   bi = 0x7fffffff;
    for (int n = tid; n < NKEYS; n += blockDim.x) {
      bool skip = false;
      for (int c = 0; c < j; ++c) skip |= (chosen_i[c] == n);
      if (skip) continue;
      float v = score_at(partial, n, nchunks);
      if (v > bv || (v == bv && n < bi)) { bv = v; bi = n; }
    }
#pragma unroll
    for (int o = 16; o > 0; o >>= 1) {
      float ov = __shfl_down(bv, o, 32);
      int   oi = __shfl_down(bi, o, 32);
      if (ov > bv || (ov == bv && oi < bi)) { bv = ov; bi = oi; }
    }
    if (lane == 0) { rv[warp] = bv; ri[warp] = bi; }
    __syncthreads();
    if (warp == 0) {
      float v2 = (lane < nw) ? rv[lane] : -3.4e38f;
      int   i2 = (lane < nw) ? ri[lane] : 0x7fffffff;
#pragma unroll
      for (int o = 16; o > 0; o >>= 1) {
        float ov = __shfl_down(v2, o, 32);
        int   oi = __shfl_down(i2, o, 32);
        if (ov > v2 || (ov == v2 && oi < i2)) { v2 = ov; i2 = oi; }
      }
      if (lane == 0) { chosen_s[j] = v2; chosen_i[j] = i2; }
    }
    __syncthreads();

    if (j == 0) {
      // softmax denominator with m = max score (just found)
      float m  = chosen_s[0] * inv;
      float ps = 0.f;
      for (int n = tid; n < NKEYS; n += blockDim.x)
        ps += expf(score_at(partial, n, nchunks) * inv - m);
#pragma unroll
      for (int o = 16; o > 0; o >>= 1) ps += __shfl_down(ps, o, 32);
      if (lane == 0) psum[warp] = ps;
      __syncthreads();
      if (tid == 0) {
        float t = 0.f;
        for (int w = 0; w < nw; ++w) t += psum[w];
        s_sum = t;
      }
      __syncthreads();
    }
  }

  // out[0 .. 10*1024) = values[idx]  ;  out[10*1024 .. +10) = top weights
  float m      = chosen_s[0] * inv;
  float invsum = 1.0f / s_sum;
  for (int e = tid; e < TOPK * DIM; e += blockDim.x) {
    int j = e >> 10;
    int c = e & (DIM - 1);
    out[e] = values[(size_t)chosen_i[j] * DIM + c];
  }
  if (tid < TOPK) out[TOPK * DIM + tid] = expf(chosen_s[tid] * inv - m) * invsum;
}

// ---------------------------------------------------------------------------
extern "C" void kernel_launch(void* const* d_in, const int* in_sizes, int n_in,
                              void* d_out, int out_size, void* d_ws, size_t ws_size,
                              hipStream_t stream) {
  const float* query  = (const float*)d_in[0];   // [1024]
  const float* keys   = (const float*)d_in[1];   // [32768,1024]
  const float* values = (const float*)d_in[2];   // [32768,1024]
  const float* Wq     = (const float*)d_in[3];   // [1024,1024]
  const float* bq     = (const float*)d_in[4];   // [1024]
  // d_in[5] = k (==10, compile-time TOPK)

  float* ws      = (float*)d_ws;
  float* qout    = ws;          // 1024 floats
  float* partial = ws + DIM;    // (1<<lg) * 32768 floats

  // Pick K-split factor by available scratch (deterministic: ws_size fixed).
  size_t need2 = (DIM + ((size_t)NKEYS << 2)) * sizeof(float);  // lg=2
  size_t need1 = (DIM + ((size_t)NKEYS << 1)) * sizeof(float);  // lg=1
  int lg = (ws_size >= need2) ? 2 : ((ws_size >= need1) ? 1 : 0);

  // 1024 rows, one wave each: 128 blocks x 256 threads
  proj_kernel<<<DIM / 8, 256, 0, stream>>>(Wq, query, bq, qout);
  // (2048 << lg) waves, 8 waves per 256-thread block
  scores_wmma_kernel<<<(256 << lg), 256, 0, stream>>>(keys, qout, partial, lg);
  // single-block top-k + softmax + gather (combines partials in fixed order)
  topk_softmax_gather_kernel<<<1, 1024, 0, stream>>>(partial, 1 << lg, values,
                                                     (float*)d_out);
}